// BiLSTMCRF_14018773254548
// MI455X (gfx1250) — compile-verified
//
#include <hip/hip_runtime.h>
#include <math.h>

// ---------------- types / constants ----------------
typedef __attribute__((ext_vector_type(2))) float vf2;
typedef __attribute__((ext_vector_type(8))) float vf8;

#define B_    256
#define L_    300
#define CD_   30
#define CDP_  32
#define HH_   64
#define H_    128
#define G4_   256           // 4*HH
#define WN_   48
#define WD_   128
#define D1P_  112           // 100 padded
#define D2P_  96            // 94 padded
#define TP_   16            // 6 padded
#define NMAP_ 304           // 300 padded
#define SLOPE_ 0.01f

__device__ __forceinline__ float leakyf(float x){ return x >= 0.f ? x : SLOPE_ * x; }
__device__ __forceinline__ float sigmf(float x){ return 1.f / (1.f + __expf(-x)); }

// ---------------- generic WMMA f32 GEMM: C = leaky(A @ W^T + bias) ----------------
// Compile-time K and tile count NT -> branch-free inner loop, no acc spills.
// A: [M, lda] row-major (cols 0..K-1 used, K % 4 == 0)
// W: [Np, ldw] row-major (B[k][n] = W[n][k])
// C: [M, ldc], columns [n0base + 64*blockIdx.y , +16*NT). Grid: (M/16, groups), block 32.
template<int K, int NT>
__global__ void k_gemm_leaky(const float* __restrict__ A, const float* __restrict__ W,
                             const float* __restrict__ bias, float* __restrict__ C,
                             int lda, int ldw, int ldc, int coloff, int n0base)
{
    const int lane = threadIdx.x;
    const int m  = lane & 15;
    const int hi = lane >> 4;
    const int kb = hi << 1;
    const int row0 = blockIdx.x << 4;
    const int n0   = n0base + (blockIdx.y << 6);

    vf8 acc[NT];
#pragma unroll
    for (int j = 0; j < NT; ++j)
#pragma unroll
        for (int r = 0; r < 8; ++r) acc[j][r] = 0.f;

    const float* arow = A + (size_t)(row0 + m) * lda;
    const float* wrow[NT];
#pragma unroll
    for (int j = 0; j < NT; ++j) wrow[j] = W + (size_t)(n0 + j * 16 + m) * ldw;

#pragma unroll 4
    for (int k = 0; k < K; k += 4) {
        if ((k & 31) == 0) __builtin_prefetch(arow + k + 64, 0, 0);   // global_prefetch_b8
        vf2 a = *(const vf2*)(arow + k + kb);
#pragma unroll
        for (int j = 0; j < NT; ++j) {
            vf2 b = *(const vf2*)(wrow[j] + k + kb);
            acc[j] = __builtin_amdgcn_wmma_f32_16x16x4_f32(
                false, a, false, b, (short)0, acc[j], false, false);
        }
    }
#pragma unroll
    for (int j = 0; j < NT; ++j) {
        int col = n0 + j * 16 + m;
        float bv = bias[col];
#pragma unroll
        for (int r = 0; r < 8; ++r) {
            int row = row0 + r + hi * 8;
            C[(size_t)row * ldc + coloff + col] = leakyf(acc[j][r] + bv);
        }
    }
}

// host-side dispatch: full 64-col groups (NT=4) + one remainder launch (NT=1..3)
template<int K>
static void gemm_launch(const float* A, const float* W, const float* bias, float* C,
                        int M, int Np, int lda, int ldw, int ldc, int coloff,
                        hipStream_t s)
{
    int full = Np >> 6;
    if (full > 0)
        k_gemm_leaky<K, 4><<<dim3(M / 16, full), 32, 0, s>>>(A, W, bias, C, lda, ldw, ldc, coloff, 0);
    int rem = (Np & 63) >> 4;
    int n0 = full << 6;
    if (rem == 1)
        k_gemm_leaky<K, 1><<<dim3(M / 16, 1), 32, 0, s>>>(A, W, bias, C, lda, ldw, ldc, coloff, n0);
    else if (rem == 2)
        k_gemm_leaky<K, 2><<<dim3(M / 16, 1), 32, 0, s>>>(A, W, bias, C, lda, ldw, ldc, coloff, n0);
    else if (rem == 3)
        k_gemm_leaky<K, 3><<<dim3(M / 16, 1), 32, 0, s>>>(A, W, bias, C, lda, ldw, ldc, coloff, n0);
}

// ---------------- BiLSTM scan with WMMA recurrence ----------------
// Grid: (B/16, 2 directions), block 128 (4 waves; wave w handles gate block w: 64 cols).
__global__ void k_lstm(const float* __restrict__ Xf, const float* __restrict__ Xb,
                       const float* __restrict__ Wih_f, const float* __restrict__ Wih_b,
                       const float* __restrict__ Whh_f, const float* __restrict__ Whh_b,
                       const float* __restrict__ bias_f, const float* __restrict__ bias_b,
                       const int* __restrict__ length, float* __restrict__ out)
{
    __shared__ float sh[16 * HH_];
    __shared__ float sc[16 * HH_];
    __shared__ float sg[16 * G4_];
    __shared__ int   slen[16];

    const int dir = blockIdx.y;
    const float* X   = dir ? Xb     : Xf;
    const float* Wih = dir ? Wih_b  : Wih_f;
    const float* Whh = dir ? Whh_b  : Whh_f;
    const float* bs  = dir ? bias_b : bias_f;

    const int tid  = threadIdx.x;
    const int wave = tid >> 5;
    const int lane = tid & 31;
    const int m  = lane & 15;
    const int hi = lane >> 4;
    const int kb = hi << 1;
    const int bb0 = blockIdx.x << 4;
    const int nbase = wave << 6;

    const float* whr[4];
    const float* wir[4];
#pragma unroll
    for (int j = 0; j < 4; ++j) {
        whr[j] = Whh + (size_t)(nbase + j * 16 + m) * HH_;
        wir[j] = Wih + (size_t)(nbase + j * 16 + m) * CDP_;
    }

    for (int i = tid; i < 16 * HH_; i += 128) { sh[i] = 0.f; sc[i] = 0.f; }
    if (tid < 16) slen[tid] = length[bb0 + tid];
    __syncthreads();

    for (int t = 0; t < L_; ++t) {
        vf8 acc[4];
#pragma unroll
        for (int j = 0; j < 4; ++j)
#pragma unroll
            for (int r = 0; r < 8; ++r) acc[j][r] = 0.f;

        // hidden contribution: K = 64 (h in LDS, W_hh from global/L2)
#pragma unroll 4
        for (int k = 0; k < HH_; k += 4) {
            vf2 a = *(const vf2*)(sh + m * HH_ + k + kb);
#pragma unroll
            for (int j = 0; j < 4; ++j) {
                vf2 b = *(const vf2*)(whr[j] + k + kb);
                acc[j] = __builtin_amdgcn_wmma_f32_16x16x4_f32(
                    false, a, false, b, (short)0, acc[j], false, false);
            }
        }
        // input contribution: K = 32 (padded char embeddings)
        const float* xr = X + ((size_t)(bb0 + m) * L_ + t) * CDP_;
#pragma unroll 4
        for (int k = 0; k < CDP_; k += 4) {
            vf2 a = *(const vf2*)(xr + k + kb);
#pragma unroll
            for (int j = 0; j < 4; ++j) {
                vf2 b = *(const vf2*)(wir[j] + k + kb);
                acc[j] = __builtin_amdgcn_wmma_f32_16x16x4_f32(
                    false, a, false, b, (short)0, acc[j], false, false);
            }
        }
        // spill gates to LDS
#pragma unroll
        for (int j = 0; j < 4; ++j)
#pragma unroll
            for (int r = 0; r < 8; ++r)
                sg[(r + hi * 8) * G4_ + nbase + j * 16 + m] = acc[j][r];
        __syncthreads();

        // elementwise cell update + masked/reversed output write
        for (int e = tid; e < 16 * HH_; e += 128) {
            int row = e >> 6, hh = e & 63;
            float gi = sg[row * G4_            + hh] + bs[hh];
            float gf = sg[row * G4_ +   HH_    + hh] + bs[HH_ + hh];
            float gg = sg[row * G4_ + 2 * HH_  + hh] + bs[2 * HH_ + hh];
            float go = sg[row * G4_ + 3 * HH_  + hh] + bs[3 * HH_ + hh];
            float c = sigmf(gf) * sc[e] + sigmf(gi) * tanhf(gg);
            float h = sigmf(go) * tanhf(c);
            sc[e] = c; sh[e] = h;
            int len = slen[row];
            if (t < len) {
                int l = dir ? (len - 1 - t) : t;
                out[((size_t)(bb0 + row) * L_ + l) * H_ + dir * HH_ + hh] = h;
            }
        }
        __syncthreads();
    }
}

// ---------------- embedding gather (+ per-sample reversed copy), K padded to 32 ----
__global__ void k_embed(const int* __restrict__ ch, const int* __restrict__ length,
                        const float* __restrict__ emb,
                        float* __restrict__ xp, float* __restrict__ rp)
{
    int idx = blockIdx.x * blockDim.x + threadIdx.x;
    if (idx >= B_ * L_) return;
    int b = idx / L_, t = idx % L_;
    int len = length[b];
    float* xo = xp + (size_t)idx * CDP_;
    int c = ch[idx];
    for (int k = 0; k < CD_; ++k) xo[k] = emb[c * CD_ + k];
    xo[30] = 0.f; xo[31] = 0.f;
    float* ro = rp + (size_t)idx * CDP_;
    if (t < len) {
        int cs = ch[b * L_ + (len - 1 - t)];
        for (int k = 0; k < CD_; ++k) ro[k] = emb[cs * CD_ + k];
    } else {
        for (int k = 0; k < CD_; ++k) ro[k] = 0.f;
    }
    ro[30] = 0.f; ro[31] = 0.f;
}

// ---------------- CNN: conv+bias+bn+leaky+maxpool fused, direct ----------------
__global__ void k_conv1(const float* __restrict__ x, const float* __restrict__ w,
                        const float* __restrict__ cb, const float* __restrict__ bs,
                        const float* __restrict__ bb, float* __restrict__ out)
{
    int idx = blockIdx.x * blockDim.x + threadIdx.x;      // [B,16,12,32]
    if (idx >= B_ * 16 * 12 * 32) return;
    int px = idx & 31, py = (idx >> 5) % 12, oc = (idx / (12 * 32)) & 15, b = idx / (16 * 12 * 32);
    const float* wi = w + oc * 9;
    float mx = -1e30f;
    for (int dy = 0; dy < 2; ++dy) for (int dx = 0; dx < 2; ++dx) {
        int oy = py * 2 + dy, ox = px * 2 + dx;
        int iy0 = oy * 2 - 1, ix0 = ox * 2 - 1;           // stride 2, pad 1
        float s = 0.f;
        for (int ky = 0; ky < 3; ++ky) { int iy = iy0 + ky; if (iy < 0 || iy >= WN_) continue;
            for (int kx = 0; kx < 3; ++kx) { int ix = ix0 + kx; if (ix < 0 || ix >= WD_) continue;
                s += x[((size_t)b * WN_ + iy) * WD_ + ix] * wi[ky * 3 + kx]; } }
        float v = leakyf((s + cb[oc]) * bs[oc] + bb[oc]);
        mx = v > mx ? v : mx;
    }
    out[idx] = mx;
}

__global__ void k_conv2(const float* __restrict__ p1, const float* __restrict__ w,
                        const float* __restrict__ cb, const float* __restrict__ bs,
                        const float* __restrict__ bb, float* __restrict__ out)
{
    int idx = blockIdx.x * blockDim.x + threadIdx.x;      // [B,32,6,16]
    if (idx >= B_ * 32 * 6 * 16) return;
    int px = idx & 15, py = (idx >> 4) % 6, oc = (idx / (6 * 16)) % 32, b = idx / (32 * 6 * 16);
    float mx = -1e30f;
    for (int dy = 0; dy < 2; ++dy) for (int dx = 0; dx < 2; ++dx) {
        int oy = py * 2 + dy, ox = px * 2 + dx;           // 12x32, pad 1, stride 1
        float s = 0.f;
        for (int ic = 0; ic < 16; ++ic) {
            const float* in = p1 + ((size_t)(b * 16 + ic) * 12) * 32;
            const float* wk = w + ((oc * 16 + ic) * 9);
            for (int ky = 0; ky < 3; ++ky) { int iy = oy + ky - 1; if (iy < 0 || iy >= 12) continue;
                for (int kx = 0; kx < 3; ++kx) { int ix = ox + kx - 1; if (ix < 0 || ix >= 32) continue;
                    s += in[iy * 32 + ix] * wk[ky * 3 + kx]; } }
        }
        float v = leakyf((s + cb[oc]) * bs[oc] + bb[oc]);
        mx = v > mx ? v : mx;
    }
    out[idx] = mx;
}

__global__ void k_conv3(const float* __restrict__ p2, const float* __restrict__ w,
                        const float* __restrict__ cb, const float* __restrict__ bs,
                        const float* __restrict__ bb, float* __restrict__ cnnA)
{
    int idx = blockIdx.x * blockDim.x + threadIdx.x;      // [B,64,3,8]
    if (idx >= B_ * 64 * 3 * 8) return;
    int px = idx & 7, py = (idx >> 3) % 3, oc = (idx / 24) % 64, b = idx / (64 * 24);
    float mx = -1e30f;
    for (int dy = 0; dy < 2; ++dy) for (int dx = 0; dx < 2; ++dx) {
        int oy = py * 2 + dy, ox = px * 2 + dx;           // 6x16, pad 1, stride 1
        float s = 0.f;
        for (int ic = 0; ic < 32; ++ic) {
            const float* in = p2 + ((size_t)(b * 32 + ic) * 6) * 16;
            const float* wk = w + ((oc * 32 + ic) * 9);
            for (int ky = 0; ky < 3; ++ky) { int iy = oy + ky - 1; if (iy < 0 || iy >= 6) continue;
                for (int kx = 0; kx < 3; ++kx) { int ix = ox + kx - 1; if (ix < 0 || ix >= 16) continue;
                    s += in[iy * 16 + ix] * wk[ky * 3 + kx]; } }
        }
        float v = leakyf((s + cb[oc]) * bs[oc] + bb[oc]);
        mx = v > mx ? v : mx;
    }
    // reshape [64,3,8] -> rows of cnn_flat [128,12], stored K-padded to 16
    int fi = oc * 24 + py * 8 + px;
    int row = fi / 12, col = fi % 12;
    cnnA[((size_t)b * 128 + row) * 16 + col] = mx;
}

// ---------------- repack cnn_mapping [B,128,304pad] -> cat cols 128..255 ----------
__global__ void k_repack(const float* __restrict__ cm, float* __restrict__ cat)
{
    int idx = blockIdx.x * blockDim.x + threadIdx.x;      // B*L*H elements
    if (idx >= B_ * L_ * H_) return;
    int b = idx / (L_ * H_);
    int r = idx % (L_ * H_);
    int l = r / H_, h = r % H_;
    int fi = l * H_ + h;                                   // flatten of [128,300]
    int i = fi / 300, j = fi % 300;
    cat[((size_t)(b * L_ + l)) * 256 + 128 + h] = cm[((size_t)(b * 128 + i)) * NMAP_ + j];
}

// ---------------- CRF: real path score + forward algorithm, per batch element ------
__global__ void k_crf(const float* __restrict__ logits, const int* __restrict__ tags,
                      const int* __restrict__ length, const float* __restrict__ trans,
                      float* __restrict__ partial)
{
    int b = blockIdx.x * blockDim.x + threadIdx.x;
    if (b >= B_) return;
    float tr[36];
    for (int i = 0; i < 36; ++i) tr[i] = trans[i];
    int len = length[b];
    const float* lg = logits + (size_t)b * L_ * TP_;
    const int* tg = tags + b * L_;

    float real = 0.f; int prev = 4;                        // START
    for (int l = 0; l < len; ++l) {
        int t = tg[l];
        real += lg[l * TP_ + t] + tr[prev * 6 + t];
        prev = t;
    }
    real += tr[tg[len - 1] * 6 + 5];                       // STOP

    float alpha[6] = {0.f, 0.f, 0.f, 0.f, 0.f, 0.f};
    for (int l = 0; l < len; ++l) {
        float na[6];
        for (int j = 0; j < 6; ++j) {
            float mx = -1e30f;
            for (int i = 0; i < 6; ++i) { float v = alpha[i] + tr[i * 6 + j]; mx = v > mx ? v : mx; }
            float s = 0.f;
            for (int i = 0; i < 6; ++i) s += __expf(alpha[i] + tr[i * 6 + j] - mx);
            na[j] = mx + __logf(s) + lg[l * TP_ + j];
        }
        for (int j = 0; j < 6; ++j) alpha[j] = na[j];
    }
    float mx = -1e30f;
    for (int j = 0; j < 6; ++j) { float v = alpha[j] + tr[j * 6 + 5]; mx = v > mx ? v : mx; }
    float s = 0.f;
    for (int j = 0; j < 6; ++j) s += __expf(alpha[j] + tr[j * 6 + 5] - mx);
    partial[b] = (mx + __logf(s)) - real;
}

__global__ void k_reduce(const float* __restrict__ partial, float* __restrict__ out)
{
    __shared__ float sm[256];
    int t = threadIdx.x;
    sm[t] = partial[t];
    __syncthreads();
    for (int s = 128; s > 0; s >>= 1) { if (t < s) sm[t] += sm[t + s]; __syncthreads(); }
    if (t == 0) out[0] = sm[0];
}

// ---------------- small prep kernels ----------------
__global__ void k_pad2d(const float* __restrict__ src, float* __restrict__ dst,
                        int N, int K, int Np, int Kp)
{
    int i = blockIdx.x * blockDim.x + threadIdx.x;
    if (i >= Np * Kp) return;
    int n = i / Kp, k = i % Kp;
    dst[i] = (n < N && k < K) ? src[n * K + k] : 0.f;
}
__global__ void k_pad1d(const float* __restrict__ src, float* __restrict__ dst, int N, int Np)
{
    int i = blockIdx.x * blockDim.x + threadIdx.x;
    if (i >= Np) return;
    dst[i] = (i < N) ? src[i] : 0.f;
}
__global__ void k_add2(const float* __restrict__ a, const float* __restrict__ b,
                       float* __restrict__ o, int n)
{
    int i = blockIdx.x * blockDim.x + threadIdx.x;
    if (i < n) o[i] = a[i] + b[i];
}
__global__ void k_fill0(float* __restrict__ p, size_t n)
{
    size_t i = (size_t)blockIdx.x * blockDim.x + threadIdx.x;
    size_t stride = (size_t)gridDim.x * blockDim.x;
    for (; i < n; i += stride) p[i] = 0.f;
}

// ---------------- host launcher ----------------
extern "C" void kernel_launch(void* const* d_in, const int* in_sizes, int n_in,
                              void* d_out, int out_size, void* d_ws, size_t ws_size,
                              hipStream_t stream)
{
    (void)in_sizes; (void)n_in; (void)out_size; (void)ws_size;
    const int*   characters = (const int*)  d_in[0];
    const int*   tags       = (const int*)  d_in[1];
    const int*   length     = (const int*)  d_in[2];
    const float* words      = (const float*)d_in[3];
    const float* emb        = (const float*)d_in[4];
    const float* w_ih_f     = (const float*)d_in[5];
    const float* w_hh_f     = (const float*)d_in[6];
    const float* b_ih_f     = (const float*)d_in[7];
    const float* b_hh_f     = (const float*)d_in[8];
    const float* w_ih_b     = (const float*)d_in[9];
    const float* w_hh_b     = (const float*)d_in[10];
    const float* b_ih_b     = (const float*)d_in[11];
    const float* b_hh_b     = (const float*)d_in[12];
    const float* w_lin_lstm = (const float*)d_in[13];
    const float* b_lin_lstm = (const float*)d_in[14];
    const float* conv1_w    = (const float*)d_in[15];
    const float* conv1_b    = (const float*)d_in[16];
    const float* bn1_s      = (const float*)d_in[17];
    const float* bn1_b      = (const float*)d_in[18];
    const float* conv2_w    = (const float*)d_in[19];
    const float* conv2_b    = (const float*)d_in[20];
    const float* bn2_s      = (const float*)d_in[21];
    const float* bn2_b      = (const float*)d_in[22];
    const float* conv3_w    = (const float*)d_in[23];
    const float* conv3_b    = (const float*)d_in[24];
    const float* bn3_s      = (const float*)d_in[25];
    const float* bn3_b      = (const float*)d_in[26];
    const float* w_lin_cnn  = (const float*)d_in[27];
    const float* b_lin_cnn  = (const float*)d_in[28];
    const float* w1         = (const float*)d_in[29];
    const float* b1         = (const float*)d_in[30];
    const float* w2         = (const float*)d_in[31];
    const float* b2         = (const float*)d_in[32];
    const float* w_tag      = (const float*)d_in[33];
    const float* b_tag      = (const float*)d_in[34];
    const float* trans      = (const float*)d_in[35];
    float* out = (float*)d_out;

    // workspace layout (floats, 256-float aligned)
    float* W = (float*)d_ws;
    size_t off = 0;
    auto alloc = [&](size_t n) { size_t r = off; off += (n + 255) & ~(size_t)255; return r; };
    const size_t ML = (size_t)B_ * L_;                     // 76800
    float* xp       = W + alloc(ML * CDP_);
    float* rp       = W + alloc(ML * CDP_);
    float* lstm_h1  = W + alloc(ML * H_);                  // lstm_out, later reused as h1 [ML x 112]
    float* cat      = W + alloc(ML * 256);
    float* h2p      = W + alloc(ML * D2P_);
    float* logitsp  = W + alloc(ML * TP_);
    float* pool1    = W + alloc((size_t)B_ * 16 * 12 * 32);
    float* pool2    = W + alloc((size_t)B_ * 32 * 6 * 16);
    float* cnnA     = W + alloc((size_t)B_ * 128 * 16);
    float* cnnmapP  = W + alloc((size_t)B_ * 128 * NMAP_);
    float* w_ihf_p  = W + alloc(256 * CDP_);
    float* w_ihb_p  = W + alloc(256 * CDP_);
    float* biasCf   = W + alloc(256);
    float* biasCb   = W + alloc(256);
    float* w1p      = W + alloc((size_t)D1P_ * 256);
    float* b1p      = W + alloc(D1P_);
    float* w2p      = W + alloc((size_t)D2P_ * D1P_);
    float* b2p      = W + alloc(D2P_);
    float* wtagp    = W + alloc((size_t)TP_ * D2P_);
    float* btagp    = W + alloc(TP_);
    float* wlcnnp   = W + alloc((size_t)NMAP_ * 16);
    float* blcnnp   = W + alloc(NMAP_);
    float* partial  = W + alloc(B_);

    auto cdiv = [](int a, int b) { return (a + b - 1) / b; };

    // --- weight/bias padding and combination ---
    k_pad2d<<<cdiv(256 * CDP_, 256), 256, 0, stream>>>(w_ih_f, w_ihf_p, 256, CD_, 256, CDP_);
    k_pad2d<<<cdiv(256 * CDP_, 256), 256, 0, stream>>>(w_ih_b, w_ihb_p, 256, CD_, 256, CDP_);
    k_add2 <<<1, 256, 0, stream>>>(b_ih_f, b_hh_f, biasCf, 256);
    k_add2 <<<1, 256, 0, stream>>>(b_ih_b, b_hh_b, biasCb, 256);
    k_pad2d<<<cdiv(D1P_ * 256, 256), 256, 0, stream>>>(w1, w1p, 100, 256, D1P_, 256);
    k_pad1d<<<1, 256, 0, stream>>>(b1, b1p, 100, D1P_);
    k_pad2d<<<cdiv(D2P_ * D1P_, 256), 256, 0, stream>>>(w2, w2p, 94, 100, D2P_, D1P_);
    k_pad1d<<<1, 256, 0, stream>>>(b2, b2p, 94, D2P_);
    k_pad2d<<<cdiv(TP_ * D2P_, 256), 256, 0, stream>>>(w_tag, wtagp, 6, 94, TP_, D2P_);
    k_pad1d<<<1, 256, 0, stream>>>(b_tag, btagp, 6, TP_);
    k_pad2d<<<cdiv(NMAP_ * 16, 256), 256, 0, stream>>>(w_lin_cnn, wlcnnp, 300, 12, NMAP_, 16);
    k_pad1d<<<2, 256, 0, stream>>>(b_lin_cnn, blcnnp, 300, NMAP_);

    // --- zero-init buffers with masked / padded writes ---
    k_fill0<<<2048, 256, 0, stream>>>(lstm_h1, ML * H_);
    k_fill0<<<512, 256, 0, stream>>>(cnnA, (size_t)B_ * 128 * 16);

    // --- embeddings (+ reversed) ---
    k_embed<<<cdiv(B_ * L_, 256), 256, 0, stream>>>(characters, length, emb, xp, rp);

    // --- BiLSTM scan (WMMA recurrence) ---
    k_lstm<<<dim3(B_ / 16, 2), 128, 0, stream>>>(xp, rp, w_ihf_p, w_ihb_p,
                                                 w_hh_f, w_hh_b, biasCf, biasCb,
                                                 length, lstm_h1);

    // --- lstm_mapping -> cat[:, 0:128] ---
    gemm_launch<H_>(lstm_h1, w_lin_lstm, b_lin_lstm, cat,
                    (int)ML, H_, H_, H_, 256, 0, stream);

    // --- CNN chain ---
    k_conv1<<<cdiv(B_ * 16 * 12 * 32, 256), 256, 0, stream>>>(words, conv1_w, conv1_b, bn1_s, bn1_b, pool1);
    k_conv2<<<cdiv(B_ * 32 * 6 * 16, 256), 256, 0, stream>>>(pool1, conv2_w, conv2_b, bn2_s, bn2_b, pool2);
    k_conv3<<<cdiv(B_ * 64 * 3 * 8, 256), 256, 0, stream>>>(pool2, conv3_w, conv3_b, bn3_s, bn3_b, cnnA);

    // --- cnn_mapping GEMM [B*128,16] @ [304,16]^T -> [B*128, 304] ---
    gemm_launch<16>(cnnA, wlcnnp, blcnnp, cnnmapP,
                    B_ * 128, NMAP_, 16, 16, NMAP_, 0, stream);

    // --- repack cnn_mapping into cat[:, 128:256] ---
    k_repack<<<cdiv(B_ * L_ * H_, 256), 256, 0, stream>>>(cnnmapP, cat);

    // --- fusion MLP ---
    gemm_launch<256>(cat, w1p, b1p, lstm_h1 /* h1 reuse */,
                     (int)ML, D1P_, 256, 256, D1P_, 0, stream);
    gemm_launch<D1P_>(lstm_h1, w2p, b2p, h2p,
                      (int)ML, D2P_, D1P_, D1P_, D2P_, 0, stream);
    gemm_launch<D2P_>(h2p, wtagp, btagp, logitsp,
                      (int)ML, TP_, D2P_, D2P_, TP_, 0, stream);

    // --- CRF + final reduction ---
    k_crf<<<8, 32, 0, stream>>>(logitsp, tags, length, trans, partial);
    k_reduce<<<1, 256, 0, stream>>>(partial, out);
}